// GCN_55155970015713
// MI455X (gfx1250) — compile-verified
//
#include <hip/hip_runtime.h>
#include <hip/hip_bf16.h>

#define N_NODES 100000
#define N_EDGES 3200000
#define F_IN    128
#define F_HID   128
#define F_OUT   64
#define BN_EPS  1e-5f

typedef float v2f __attribute__((ext_vector_type(2)));
typedef float v8f __attribute__((ext_vector_type(8)));

// ---------------------------------------------------------------------------
// init kernels
// ---------------------------------------------------------------------------
__global__ void k_zero_f32(float* __restrict__ p, int n) {
    int i = blockIdx.x * blockDim.x + threadIdx.x;
    if (i < n) p[i] = 0.0f;
}

// deg = 1 (self loop), stats region = 0
__global__ void k_init_deg_stats(float* __restrict__ deg, float* __restrict__ stats) {
    int i = blockIdx.x * blockDim.x + threadIdx.x;
    if (i < N_NODES) deg[i] = 1.0f;
    if (i < 512)     stats[i] = 0.0f;
}

// out[r, f] = b2[f]  (scatter2 accumulates on top of the bias)
__global__ void k_init_out_bias(float* __restrict__ out, const float* __restrict__ b2) {
    int i = blockIdx.x * blockDim.x + threadIdx.x;
    if (i < N_NODES * F_OUT) out[i] = b2[i & (F_OUT - 1)];
}

// ---------------------------------------------------------------------------
// degree / norm
// ---------------------------------------------------------------------------
__global__ void k_degree(const int* __restrict__ dst, float* __restrict__ deg) {
    int e = blockIdx.x * blockDim.x + threadIdx.x;
    if (e < N_EDGES) atomicAdd(&deg[dst[e]], 1.0f);
}

__global__ void k_rsqrt(const float* __restrict__ deg, float* __restrict__ dinv) {
    int i = blockIdx.x * blockDim.x + threadIdx.x;
    if (i < N_NODES) dinv[i] = rsqrtf(deg[i]);
}

// ---------------------------------------------------------------------------
// f32 WMMA GEMM:  H[N_NODES, NC] = A[N_NODES, 128] * W[128, NC]
// one wave per 16x16 output tile; blockDim = 32 * (NC/16); grid = N_NODES/16
// ISA 32-bit layouts (wave32):
//   A 16x4 : lanes 0-15 -> M=lane, K = k0 + {0,1}; lanes 16-31 -> K = k0 + {2,3}
//   B 4x16 : lanes 0-15 -> N=lane, K = k0 + {0,1}; lanes 16-31 -> K = k0 + {2,3}
//   C/D    : vgpr i, lane l -> M = i + 8*(l>>4), N = l&15
// ---------------------------------------------------------------------------
template <int NC>
__global__ void k_gemm_wmma(const float* __restrict__ A,
                            const float* __restrict__ W,
                            float* __restrict__ H) {
    const int lane = threadIdx.x & 31;
    const int wave = threadIdx.x >> 5;
    const int half = lane >> 4;     // 0: K 0..1, 1: K 2..3
    const int m16  = lane & 15;     // A row within tile / B column within tile

    const int row0 = blockIdx.x * 16;       // N_NODES % 16 == 0 -> no guard
    const int col0 = wave * 16;             // NC / 16 waves per block

    v8f acc = {};

    const float* __restrict__ aRow = A + (size_t)(row0 + m16) * F_HID + 2 * half;
    const float* __restrict__ wCol = W + (size_t)(2 * half) * NC + col0 + m16;

    #pragma unroll 4
    for (int k0 = 0; k0 < F_HID; k0 += 4) {
        v2f a = *(const v2f*)(aRow + k0);                  // A[m16][k0+2h .. +1]
        v2f b;
        b[0] = wCol[(size_t)k0 * NC];                      // W[k0+2h  ][col0+m16]
        b[1] = wCol[(size_t)k0 * NC + NC];                 // W[k0+2h+1][col0+m16]
        acc = __builtin_amdgcn_wmma_f32_16x16x4_f32(
                  false, a, false, b, (short)0, acc, false, false);
    }

    float* __restrict__ o = H + (size_t)(row0 + 8 * half) * NC + col0 + m16;
    #pragma unroll
    for (int i = 0; i < 8; ++i)
        o[(size_t)i * NC] = acc[i];
}

// ---------------------------------------------------------------------------
// edge scatter: agg[dst] += h[src] * dinv[src]*dinv[dst]
// one wave per edge; F feats -> F/32 floats per lane
// edges e in [0, N_EDGES + N_NODES); tail entries are self loops
// ---------------------------------------------------------------------------
__global__ void k_scatter128(const int* __restrict__ src, const int* __restrict__ dst,
                             const float* __restrict__ dinv,
                             const float* __restrict__ h, float* __restrict__ agg) {
    const long long e = (long long)blockIdx.x * (blockDim.x >> 5) + (threadIdx.x >> 5);
    if (e >= (long long)N_EDGES + N_NODES) return;
    int s, d;
    if (e < N_EDGES) { s = src[e]; d = dst[e]; }
    else             { s = d = (int)(e - N_EDGES); }
    const float norm = dinv[s] * dinv[d];
    const int lane = threadIdx.x & 31;

    float4 v = ((const float4*)(h + (size_t)s * F_HID))[lane];
    float* o = agg + (size_t)d * F_HID + lane * 4;
    atomicAdd(o + 0, v.x * norm);
    atomicAdd(o + 1, v.y * norm);
    atomicAdd(o + 2, v.z * norm);
    atomicAdd(o + 3, v.w * norm);
}

__global__ void k_scatter64(const int* __restrict__ src, const int* __restrict__ dst,
                            const float* __restrict__ dinv,
                            const float* __restrict__ g, float* __restrict__ out) {
    const long long e = (long long)blockIdx.x * (blockDim.x >> 5) + (threadIdx.x >> 5);
    if (e >= (long long)N_EDGES + N_NODES) return;
    int s, d;
    if (e < N_EDGES) { s = src[e]; d = dst[e]; }
    else             { s = d = (int)(e - N_EDGES); }
    const float norm = dinv[s] * dinv[d];
    const int lane = threadIdx.x & 31;

    float2 v = ((const float2*)(g + (size_t)s * F_OUT))[lane];
    float* o = out + (size_t)d * F_OUT + lane * 2;
    atomicAdd(o + 0, v.x * norm);
    atomicAdd(o + 1, v.y * norm);
}

// ---------------------------------------------------------------------------
// BatchNorm: column sums / sums-of-squares of (agg + b1)
// 256 threads: thread t -> feature t&127, row parity t>>7; LDS reduce, then
// one atomicAdd per feature per block. stats[0..127]=sum, [128..255]=sumsq
// ---------------------------------------------------------------------------
__global__ void k_bn_stats(const float* __restrict__ agg, const float* __restrict__ b1,
                           float* __restrict__ stats) {
    __shared__ float s_sum[F_HID];
    __shared__ float s_sq[F_HID];
    const int f  = threadIdx.x & (F_HID - 1);
    const int rh = threadIdx.x >> 7;
    const float bias = b1[f];

    float sum = 0.0f, sq = 0.0f;
    for (int r = blockIdx.x * 2 + rh; r < N_NODES; r += gridDim.x * 2) {
        float v = agg[(size_t)r * F_HID + f] + bias;
        sum += v;
        sq  += v * v;
    }
    if (rh == 1) { s_sum[f] = sum; s_sq[f] = sq; }
    __syncthreads();
    if (rh == 0) {
        sum += s_sum[f];
        sq  += s_sq[f];
        atomicAdd(&stats[f], sum);
        atomicAdd(&stats[F_HID + f], sq);
    }
}

// scale = gamma * rsqrt(var+eps); shift = (b1 - mean)*scale + beta
__global__ void k_bn_finalize(const float* __restrict__ stats,
                              const float* __restrict__ b1,
                              const float* __restrict__ gamma,
                              const float* __restrict__ beta,
                              float* __restrict__ scale_shift) {
    const int f = threadIdx.x;
    const float inv_n = 1.0f / (float)N_NODES;
    float mean = stats[f] * inv_n;
    float var  = stats[F_HID + f] * inv_n - mean * mean;
    float sc   = gamma[f] * rsqrtf(var + BN_EPS);
    scale_shift[f]         = sc;
    scale_shift[F_HID + f] = (b1[f] - mean) * sc + beta[f];
}

// h2 = relu(agg * scale + shift), float4-vectorized
__global__ void k_bn_relu(const float* __restrict__ agg,
                          const float* __restrict__ scale_shift,
                          float* __restrict__ h2) {
    const long long i = (long long)blockIdx.x * blockDim.x + threadIdx.x;  // float4 idx
    if (i >= (long long)N_NODES * (F_HID / 4)) return;
    const int f = ((int)i & (F_HID / 4 - 1)) * 4;
    const float* sc = scale_shift;
    const float* sh = scale_shift + F_HID;
    float4 v = ((const float4*)agg)[i];
    v.x = fmaxf(0.0f, v.x * sc[f + 0] + sh[f + 0]);
    v.y = fmaxf(0.0f, v.y * sc[f + 1] + sh[f + 1]);
    v.z = fmaxf(0.0f, v.z * sc[f + 2] + sh[f + 2]);
    v.w = fmaxf(0.0f, v.w * sc[f + 3] + sh[f + 3]);
    ((float4*)h2)[i] = v;
}

// ---------------------------------------------------------------------------
// launch
// ---------------------------------------------------------------------------
extern "C" void kernel_launch(void* const* d_in, const int* in_sizes, int n_in,
                              void* d_out, int out_size, void* d_ws, size_t ws_size,
                              hipStream_t stream) {
    const float* x     = (const float*)d_in[0];
    const int*   ei    = (const int*)d_in[1];           // [2, N_EDGES]
    const float* W1    = (const float*)d_in[2];
    const float* b1    = (const float*)d_in[3];
    const float* gamma = (const float*)d_in[4];
    const float* beta  = (const float*)d_in[5];
    const float* W2    = (const float*)d_in[6];
    const float* b2    = (const float*)d_in[7];
    float*       out   = (float*)d_out;

    const int* src = ei;
    const int* dst = ei + N_EDGES;

    // workspace layout (floats)
    float* wsf   = (float*)d_ws;
    float* stats = wsf;                       // 256: sum, sumsq
    float* ssh   = wsf + 256;                 // 256: scale[128], shift[128]
    float* deg   = wsf + 512;                 // N_NODES
    float* dinv  = deg + N_NODES;             // N_NODES
    float* h     = dinv + N_NODES;            // N_NODES * 128 (reused as h2)
    float* agg   = h + (size_t)N_NODES * F_HID; // N_NODES * 128 (reused as g)

    const long long TE = (long long)N_EDGES + N_NODES;  // edges incl. self loops

    // --- init ---
    k_zero_f32<<<(N_NODES * F_HID + 255) / 256, 256, 0, stream>>>(agg, N_NODES * F_HID);
    k_init_deg_stats<<<(N_NODES + 255) / 256, 256, 0, stream>>>(deg, stats);
    k_init_out_bias<<<(N_NODES * F_OUT + 255) / 256, 256, 0, stream>>>(out, b2);

    // --- degree + norm ---
    k_degree<<<(N_EDGES + 255) / 256, 256, 0, stream>>>(dst, deg);
    k_rsqrt<<<(N_NODES + 255) / 256, 256, 0, stream>>>(deg, dinv);

    // --- conv1: h = x @ W1 (WMMA), then normalized scatter-add ---
    k_gemm_wmma<F_HID><<<N_NODES / 16, 32 * (F_HID / 16), 0, stream>>>(x, W1, h);
    {
        int wavesPerBlock = 256 / 32;
        int blocks = (int)((TE + wavesPerBlock - 1) / wavesPerBlock);
        k_scatter128<<<blocks, 256, 0, stream>>>(src, dst, dinv, h, agg);
    }

    // --- BatchNorm (+bias) + ReLU ---
    k_bn_stats<<<512, 256, 0, stream>>>(agg, b1, stats);
    k_bn_finalize<<<1, F_HID, 0, stream>>>(stats, b1, gamma, beta, ssh);
    {
        long long n4 = (long long)N_NODES * (F_HID / 4);
        k_bn_relu<<<(int)((n4 + 255) / 256), 256, 0, stream>>>(agg, ssh, h);  // h <- h2
    }

    // --- conv2: g = h2 @ W2 (WMMA), scatter into out (pre-seeded with b2) ---
    float* g = agg;  // reuse
    k_gemm_wmma<F_OUT><<<N_NODES / 16, 32 * (F_OUT / 16), 0, stream>>>(h, W2, g);
    {
        int wavesPerBlock = 256 / 32;
        int blocks = (int)((TE + wavesPerBlock - 1) / wavesPerBlock);
        k_scatter64<<<blocks, 256, 0, stream>>>(src, dst, dinv, g, out);
    }
}